// Aty_76106820485735
// MI455X (gfx1250) — compile-verified
//
#include <hip/hip_runtime.h>
#include <math.h>

typedef float v2f __attribute__((ext_vector_type(2)));
typedef float v8f __attribute__((ext_vector_type(8)));
typedef unsigned int v4u __attribute__((ext_vector_type(4)));
typedef unsigned int v8u __attribute__((ext_vector_type(8)));

#define WMMA_F32(A_, B_, C_) \
  __builtin_amdgcn_wmma_f32_16x16x4_f32(false, (A_), false, (B_), (short)0, (C_), false, false)

constexpr int Bn = 4, Cn = 16, Xn = 256, Yn = 256, Tn = 256;
constexpr int PITCH = 260;  // LDS row pitch (words): 260 mod 64 == 4 -> conflict-free b64 frags
constexpr float TWO_PI_F = 6.28318530717958647692f;

// ---------------------------------------------------------------------------
// Tensor Data Mover descriptor builders (CDNA5 D#, CDNA5 ISA).
// Group0: [1:0]=count=1, [63:32]=lds_addr, [120:64]=global_addr, [127:126]=type=2
// Group1: [17:16]=data_size(2->4B), [20]=pad_enable, [24:22]=pad_interval,
//         [31:25]=pad_amount, [79:48]=tensor_dim0, [111:80]=tensor_dim1,
//         [127:112]=tile_dim0, [143:128]=tile_dim1, [207:160]=tensor_dim0_stride
// ---------------------------------------------------------------------------
__device__ __forceinline__ v4u tdm_group0(const void* gptr, void* lptr) {
  unsigned long long ga = (unsigned long long)gptr;
  unsigned int lds = (unsigned int)(unsigned long long)lptr;  // LDS aperture: low 32 bits
  v4u g0;
  g0.x = 1u;                                                  // count=1, user mode
  g0.y = lds;
  g0.z = (unsigned int)ga;
  g0.w = ((unsigned int)(ga >> 32) & 0x01FFFFFFu) | (2u << 30);  // addr[56:32] | type=2
  return g0;
}

__device__ __forceinline__ v8u tdm_group1(unsigned int dim0, unsigned int dim1,
                                          unsigned int tile0, unsigned int tile1,
                                          unsigned int stride0, bool pad) {
  v8u g1;
  // data_size=2 (4B); optional pad: interval code 7 (=256 DWORDs), amount code 3 (=4 DWORDs)
  g1.s0 = (2u << 16) | (pad ? ((1u << 20) | (7u << 22) | (3u << 25)) : 0u);
  g1.s1 = (dim0 & 0xFFFFu) << 16;                       // [47:32]=0, [63:48]=dim0.lo
  g1.s2 = (dim0 >> 16) | ((dim1 & 0xFFFFu) << 16);      // dim0.hi | dim1.lo
  g1.s3 = (dim1 >> 16) | (tile0 << 16);                 // dim1.hi | tile_dim0
  g1.s4 = tile1;                                        // tile_dim1 | tile_dim2=0
  g1.s5 = stride0;                                      // tensor_dim0_stride[31:0]
  g1.s6 = 0u;                                           // stride0.hi | stride1.lo
  g1.s7 = 0u;
  return g1;
}

__device__ __forceinline__ void tdm_load_to_lds(v4u g0, v8u g1) {
  asm volatile("tensor_load_to_lds %0, %1" ::"s"(g0), "s"(g1) : "memory");
}

// ---------------------------------------------------------------------------
// K0: transposed DFT twiddles with (-1)^x folded in.
// ET[x][m] = (-1)^x * exp(+2*pi*i * x*m / 256); exact integer phase reduction.
// ---------------------------------------------------------------------------
__global__ __launch_bounds__(256) void twiddle_kernel(float* __restrict__ ETr,
                                                      float* __restrict__ ETi) {
  int x = blockIdx.x;
  int m = threadIdx.x;
  int p = (x * m) & 255;
  float ang = (float)p * (TWO_PI_F / 256.0f);
  float s, c;
  __sincosf(ang, &s, &c);
  float sgn = (x & 1) ? -1.0f : 1.0f;
  ETr[x * 256 + m] = sgn * c;
  ETi[x * 256 + m] = sgn * s;
}

// ---------------------------------------------------------------------------
// K1: centered IFFT along X as a complex GEMM (M=x', N=t, K=x), per (b,c).
// Ks[b,x',c,t] = mask[b,x',t] * ((-1)^{x'}/256) * sum_x ET[x][x'] * kdata[b,c,x,t]
// kdata slab (256x x 16t, row stride 1KB) staged to LDS by the TDM.
// Output layout [b][x'][c][t]: D-tile stores are contiguous over t.
// ---------------------------------------------------------------------------
__global__ __launch_bounds__(256) void dft_kernel(
    const float* __restrict__ kr, const float* __restrict__ ki,
    const float* __restrict__ mask,
    const float* __restrict__ ETr, const float* __restrict__ ETi,
    float* __restrict__ Ksr, float* __restrict__ Ksi) {
  __shared__ float sKr[256 * 16];
  __shared__ float sKi[256 * 16];

  int bid = blockIdx.x;
  int tT = bid & 15;
  int c  = (bid >> 4) & 15;
  int b  = bid >> 8;
  int t0 = tT * 16;
  int tid = threadIdx.x;

  const float* kr_bc = kr + ((size_t)(b * Cn + c)) * Xn * Yn;
  const float* ki_bc = ki + ((size_t)(b * Cn + c)) * Xn * Yn;

  // TDM: 2D tile, 256 rows x 16 floats, global row stride 256 floats -> LDS [x][t].
  if ((tid >> 5) == 0) {
    v8u g1 = tdm_group1(/*dim0=*/16, /*dim1=*/256, /*tile0=*/16, /*tile1=*/256,
                        /*stride0=*/256, /*pad=*/false);
    tdm_load_to_lds(tdm_group0(kr_bc + t0, sKr), g1);
    tdm_load_to_lds(tdm_group0(ki_bc + t0, sKi), g1);
    __builtin_amdgcn_s_wait_tensorcnt(0);
  }
  __syncthreads();

  int lane  = tid & 31;
  int wave  = tid >> 5;
  int lm    = lane & 15;
  int khalf = (lane >> 4) << 1;  // 0 or 2

  for (int ti = 0; ti < 2; ++ti) {
    int xT = wave + ti * 8;        // x'-tile index 0..15
    int m  = xT * 16 + lm;         // A-matrix row = x'
    v8f accRp = {}, accRm = {}, accI = {};

#pragma unroll 4
    for (int k0 = 0; k0 < 256; k0 += 4) {
      int kb = k0 + khalf;
      v2f ar, ai, br, bi;
      ar.x = ETr[kb * 256 + m];        ar.y = ETr[(kb + 1) * 256 + m];
      ai.x = ETi[kb * 256 + m];        ai.y = ETi[(kb + 1) * 256 + m];
      br.x = sKr[kb * 16 + lm];        br.y = sKr[(kb + 1) * 16 + lm];
      bi.x = sKi[kb * 16 + lm];        bi.y = sKi[(kb + 1) * 16 + lm];
      accRp = WMMA_F32(ar, br, accRp);   // + Ar*Br
      accRm = WMMA_F32(ai, bi, accRm);   // + Ai*Bi  (subtract later)
      accI  = WMMA_F32(ar, bi, accI);    // + Ar*Bi
      accI  = WMMA_F32(ai, br, accI);    // + Ai*Br
    }

    // D layout: lane -> N = t (lm); VGPR j -> M = j + (lane>=16 ? 8 : 0)
    int t    = t0 + lm;
    int mofs = (lane >> 4) * 8;
    for (int j = 0; j < 8; ++j) {
      int xp  = xT * 16 + j + mofs;
      float sc = ((xp & 1) ? -1.0f : 1.0f) * (1.0f / 256.0f);
      float mk = mask[((size_t)(b * Xn + xp)) * Yn + t];  // mask[b,x',t] ({0,1}: mask^2==mask)
      float dr = sc * mk * (accRp[j] - accRm[j]);
      float di = sc * mk * accI[j];
      // [b][x'][c][t]: contiguous over t across lanes -> coalesced 64B stores
      size_t o = (((size_t)(b * Xn + xp)) * Cn + c) * Tn + t;
      Ksr[o] = dr;
      Ksi[o] = di;
    }
  }
}

// ---------------------------------------------------------------------------
// K2: adjoint time-segmented encode + coil combine, per (b,x).
// ci[y,c] = sum_t conj(fmt[t,y]) * e^{-i*2pi*t_eff[t]*field[b,x,y]} * Ks[b,x,c,t]
// rec[b,x,y] = sum_c ci[y,c] * conj(csm[b,c,x,y])
// Ks slab (16c x 256t) staged by TDM with pad_enable: +4 DWORDs per 256-DWORD
// row -> LDS pitch 260 words, bank-conflict-free ds_load_b64 B-fragments.
// ---------------------------------------------------------------------------
__global__ __launch_bounds__(256) void recon_kernel(
    const float* __restrict__ Ksr, const float* __restrict__ Ksi,
    const float* __restrict__ csr, const float* __restrict__ csi,
    const float* __restrict__ fmtr, const float* __restrict__ fmti,
    const float* __restrict__ field, const float* __restrict__ tl,
    const unsigned char* __restrict__ updown,
    float2* __restrict__ out) {
  __shared__ float sKr[Cn * PITCH];
  __shared__ float sKi[Cn * PITCH];

  int bid = blockIdx.x;
  int x = bid & 255;
  int b = bid >> 8;
  int tid = threadIdx.x;

  const float* Ksr_bx = Ksr + ((size_t)(b * Xn + x)) * Cn * Tn;
  const float* Ksi_bx = Ksi + ((size_t)(b * Xn + x)) * Cn * Tn;

  // TDM: 16 rows x 256 floats, contiguous in global; pad inserts 4 DWORDs per row.
  if ((tid >> 5) == 0) {
    v8u g1 = tdm_group1(/*dim0=*/256, /*dim1=*/16, /*tile0=*/256, /*tile1=*/16,
                        /*stride0=*/256, /*pad=*/true);
    tdm_load_to_lds(tdm_group0(Ksr_bx, sKr), g1);
    tdm_load_to_lds(tdm_group0(Ksi_bx, sKi), g1);
    __builtin_amdgcn_s_wait_tensorcnt(0);
  }
  __syncthreads();

  bool up   = updown[0] != 0;
  int lane  = tid & 31;
  int wave  = tid >> 5;
  int lm    = lane & 15;
  int khalf = (lane >> 4) << 1;

  for (int ti = 0; ti < 2; ++ti) {
    int yT = wave + ti * 8;
    int y  = yT * 16 + lm;  // A-matrix row = y
    float w = TWO_PI_F * field[((size_t)(b * Xn + x)) * Yn + y];
    v8f accRp = {}, accRm = {}, accI = {};

#pragma unroll 4
    for (int k0 = 0; k0 < 256; k0 += 4) {
      int ta = k0 + khalf;  // even -> 8B-aligned b64 LDS fragment loads
      int tb = ta + 1;
      float tea = tl[up ? ta : (Tn - 1 - ta)];
      float teb = tl[up ? tb : (Tn - 1 - tb)];
      float sa, ca, sb, cb;
      __sincosf(w * tea, &sa, &ca);
      __sincosf(w * teb, &sb, &cb);
      float fra = fmtr[ta * Yn + y], fia = fmti[ta * Yn + y];
      float frb = fmtr[tb * Yn + y], fib = fmti[tb * Yn + y];
      v2f ar, ai;
      // W = conj(fmt) * (cos - i sin)
      ar.x = fra * ca - fia * sa;   ai.x = -(fra * sa + fia * ca);
      ar.y = frb * cb - fib * sb;   ai.y = -(frb * sb + fib * cb);
      v2f br = *(const v2f*)&sKr[lm * PITCH + ta];   // B[k][n=c]: one ds_load_b64
      v2f bi = *(const v2f*)&sKi[lm * PITCH + ta];
      accRp = WMMA_F32(ar, br, accRp);
      accRm = WMMA_F32(ai, bi, accRm);
      accI  = WMMA_F32(ar, bi, accI);
      accI  = WMMA_F32(ai, br, accI);
    }

    // D layout: lane -> N = c (lm); VGPR j -> M = j + (lane>=16 ? 8 : 0)
    int cidx = lm;
    int mofs = (lane >> 4) * 8;
    for (int j = 0; j < 8; ++j) {
      int yo = yT * 16 + j + mofs;
      float dr = accRp[j] - accRm[j];
      float di = accI[j];
      size_t co = (((size_t)(b * Cn + cidx)) * Xn + x) * Yn + yo;
      float cr = csr[co], cim = csi[co];
      // (dr + i di) * conj(cr + i cim)
      float vr = dr * cr + di * cim;
      float vi = di * cr - dr * cim;
      // reduce over the 16 coil lanes (stays within each half-wave)
      for (int ms = 1; ms < 16; ms <<= 1) {
        vr += __shfl_xor(vr, ms, 32);
        vi += __shfl_xor(vi, ms, 32);
      }
      if (lm == 0) {
        out[((size_t)(b * Xn + x)) * Yn + yo] = make_float2(vr, vi);
      }
    }
  }
}

// ---------------------------------------------------------------------------
// Launch
// ---------------------------------------------------------------------------
extern "C" void kernel_launch(void* const* d_in, const int* in_sizes, int n_in,
                              void* d_out, int out_size, void* d_ws, size_t ws_size,
                              hipStream_t stream) {
  const float* kdata_r = (const float*)d_in[0];
  const float* kdata_i = (const float*)d_in[1];
  const float* csm_r   = (const float*)d_in[2];
  const float* csm_i   = (const float*)d_in[3];
  const float* mask    = (const float*)d_in[4];
  const float* field   = (const float*)d_in[5];
  const float* fmt_r   = (const float*)d_in[6];
  const float* fmt_i   = (const float*)d_in[7];
  const float* tl      = (const float*)d_in[8];
  const unsigned char* updown = (const unsigned char*)d_in[9];

  char* ws = (char*)d_ws;
  const size_t ET_ELEMS = (size_t)Xn * Xn;                 // 65536
  const size_t KS_ELEMS = (size_t)Bn * Xn * Cn * Tn;       // 4194304
  float* ETr = (float*)ws;
  float* ETi = ETr + ET_ELEMS;
  float* Ksr = ETi + ET_ELEMS;
  float* Ksi = Ksr + KS_ELEMS;
  // total ws use: (2*65536 + 2*4194304) * 4 B = ~34 MB

  twiddle_kernel<<<Xn, 256, 0, stream>>>(ETr, ETi);
  dft_kernel<<<Bn * Cn * (Tn / 16), 256, 0, stream>>>(kdata_r, kdata_i, mask,
                                                      ETr, ETi, Ksr, Ksi);
  recon_kernel<<<Bn * Xn, 256, 0, stream>>>(Ksr, Ksi, csm_r, csm_i, fmt_r, fmt_i,
                                            field, tl, updown, (float2*)d_out);
}